// QuanvolutionHybrid_65481071405432
// MI455X (gfx1250) — compile-verified
//
#include <hip/hip_runtime.h>
#include <hip/hip_bf16.h>

// Fused Quanvolution hybrid for gfx1250 (MI455X, wave32).
//   conv2x2/s2 -> 4-qubit statevector filter (algebraically reduced) -> FC via
//   V_WMMA_F32_16X16X4_F32 (exact fp32) -> log_softmax.  One kernel, features
//   staged in LDS (never touch HBM), 8-wave split-K GEMM per 16-row tile,
//   2-deep WMMA software pipelining (dual accumulators).

typedef float v2f __attribute__((ext_vector_type(2)));
typedef float v8f __attribute__((ext_vector_type(8)));

#define NPAT 196   // 14*14 patches per image
#define KDIM 784   // 4 channels * 196 = FC input dim
#define LDSF 788   // padded LDS row stride (row*788 % 64 banks -> conflict-free)

// RY(theta) butterfly on qubit bitmask m (state index: wire0 = bit3 ... wire3 = bit0)
__device__ __forceinline__ void qry(float* s, int m, float c, float sn) {
#pragma unroll
  for (int i = 0; i < 16; ++i) {
    if ((i & m) == 0) {
      float a = s[i], b = s[i | m];
      s[i]     = c * a - sn * b;
      s[i | m] = sn * a + c * b;
    }
  }
}

// CNOT: if ctrl bit set, flip tgt bit (pure permutation -> register renames)
__device__ __forceinline__ void qcnot(float* s, int mc, int mt) {
#pragma unroll
  for (int i = 0; i < 16; ++i) {
    if ((i & mc) != 0 && (i & mt) == 0) {
      float t = s[i]; s[i] = s[i | mt]; s[i | mt] = t;
    }
  }
}

__global__ __launch_bounds__(256) void quanv_fused_kernel(
    const float* __restrict__ x,       // [B,1,28,28]
    const float* __restrict__ conv_w,  // [4,1,2,2]
    const float* __restrict__ conv_b,  // [4]
    const float* __restrict__ angles,  // [5]
    const float* __restrict__ fc_w,    // [10,784]
    const float* __restrict__ fc_b,    // [10]
    float* __restrict__ out,           // [B,10]
    int B)
{
  __shared__ float lds_feats[16 * LDSF];   // 16 rows x 784 features (padded)
  __shared__ float lds_part[8 * 256];      // 8 waves' partial C tiles
  __shared__ float lds_tile[16 * 16];      // reduced logits tile

  const int tid  = threadIdx.x;
  const int row0 = blockIdx.x * 16;

  // ---- per-thread constant preload (all L2/L1 resident) ----
  float cw[16], cb[4], ang[5];
#pragma unroll
  for (int i = 0; i < 16; ++i) cw[i] = conv_w[i];
#pragma unroll
  for (int i = 0; i < 4; ++i)  cb[i] = conv_b[i];
#pragma unroll
  for (int i = 0; i < 5; ++i)  ang[i] = angles[i];
  float s4, c4;
  __sincosf(0.5f * ang[4], &s4, &c4);

  // ---- stage 1: conv + reduced quantum circuit -> LDS features ----
  // Circuit identity: RY(a)RY(b)=RY(a+b) on the same wire, and the fixed RYs
  // on wires 1..3 commute past earlier CNOTs on disjoint wires, so:
  //   product-state(x_i + a_i) ; CNOT(0,1) ; CNOT(1,2) ; CNOT(2,3) ; RY(a4, w0)
  for (int idx = tid; idx < 16 * NPAT; idx += 256) {
    const int row = idx / NPAT;
    const int p   = idx - row * NPAT;
    const int b   = row0 + row;
    if (b < B) {
      const int h = p / 14, w = p - h * 14;
      const float* xb = x + (size_t)b * 784 + (2 * h) * 28 + 2 * w;
      const float2 r0 = *(const float2*)(xb);        // 8B-aligned: b64 load
      const float2 r1 = *(const float2*)(xb + 28);

      float cth[4], sth[4];
#pragma unroll
      for (int c = 0; c < 4; ++c) {
        float yc = cw[c * 4 + 0] * r0.x + cw[c * 4 + 1] * r0.y +
                   cw[c * 4 + 2] * r1.x + cw[c * 4 + 3] * r1.y + cb[c];
        __sincosf(0.5f * (yc + ang[c]), &sth[c], &cth[c]);
      }
      // product state: s[q0q1q2q3] = f0(q0) f1(q1) f2(q2) f3(q3)
      float s[16];
      float q01[4];
      q01[0] = cth[0] * cth[1]; q01[1] = cth[0] * sth[1];
      q01[2] = sth[0] * cth[1]; q01[3] = sth[0] * sth[1];
#pragma unroll
      for (int i = 0; i < 4; ++i) {
        float a0 = q01[i] * cth[2], a1 = q01[i] * sth[2];
        s[i * 4 + 0] = a0 * cth[3];
        s[i * 4 + 1] = a0 * sth[3];
        s[i * 4 + 2] = a1 * cth[3];
        s[i * 4 + 3] = a1 * sth[3];
      }
      qcnot(s, 8, 4);     // CNOT(0,1)
      qcnot(s, 4, 2);     // CNOT(1,2)
      qcnot(s, 2, 1);     // CNOT(2,3)
      qry(s, 8, c4, s4);  // RY(a4) on wire 0

#pragma unroll
      for (int c = 0; c < 4; ++c) {
        const int m = 8 >> c;
        float e = 0.f;
#pragma unroll
        for (int i = 0; i < 16; ++i) {
          float pp = s[i] * s[i];
          e += (i & m) ? -pp : pp;
        }
        // feature index = c*196 + p (matches transpose(0,2,1).reshape in ref)
        lds_feats[row * LDSF + c * NPAT + p] = e;
      }
    } else {
#pragma unroll
      for (int c = 0; c < 4; ++c) lds_feats[row * LDSF + c * NPAT + p] = 0.f;
    }
  }
  __syncthreads();

  // ---- stage 2: FC via V_WMMA_F32_16X16X4_F32, 8-wave split-K ----
  // A frag (16x4 f32): lanes 0-15 -> K{0,1}, lanes 16-31 -> K{2,3}, row=lane&15
  // B frag mirrored with lane = N column.  N padding (10..15) is done with a
  // clamped in-bounds row + 0/1 mask multiply: branch-free, exec stays all-1s.
  const int wave = tid >> 5;
  const int lane = tid & 31;
  const int mrow = lane & 15;
  const int ksel = (lane >> 4) << 1;
  const int   nclamp = (mrow < 10) ? mrow : 9;
  const float nmask  = (mrow < 10) ? 1.f : 0.f;
  const float* __restrict__ bptr = fc_w + nclamp * KDIM + ksel;
  const float* aptr = &lds_feats[mrow * LDSF + ksel];

  v8f acc0 = {}, acc1 = {};
  int t = wave;
  for (; t + 8 < NPAT; t += 16) {        // dual-issue: two independent chains
    const int k0 = t * 4, k1 = k0 + 32;
    v2f a0, b0, a1, b1;
    a0[0] = aptr[k0]; a0[1] = aptr[k0 + 1];
    a1[0] = aptr[k1]; a1[1] = aptr[k1 + 1];
    b0[0] = nmask * bptr[k0]; b0[1] = nmask * bptr[k0 + 1];
    b1[0] = nmask * bptr[k1]; b1[1] = nmask * bptr[k1 + 1];
    acc0 = __builtin_amdgcn_wmma_f32_16x16x4_f32(false, a0, false, b0,
                                                 (short)0, acc0, false, false);
    acc1 = __builtin_amdgcn_wmma_f32_16x16x4_f32(false, a1, false, b1,
                                                 (short)0, acc1, false, false);
  }
  if (t < NPAT) {                        // tail (waves 0..3 have 25 steps)
    const int k0 = t * 4;
    v2f a0, b0;
    a0[0] = aptr[k0]; a0[1] = aptr[k0 + 1];
    b0[0] = nmask * bptr[k0]; b0[1] = nmask * bptr[k0 + 1];
    acc0 = __builtin_amdgcn_wmma_f32_16x16x4_f32(false, a0, false, b0,
                                                 (short)0, acc0, false, false);
  }
#pragma unroll
  for (int j = 0; j < 8; ++j)
    lds_part[wave * 256 + j * 32 + lane] = acc0[j] + acc1[j];
  __syncthreads();

  // reduce 8 partial tiles; C/D layout: VGPR j holds M=j (lanes<16) / M=8+j
  {
    const int j = tid >> 5;
    const int l = tid & 31;
    float sum = 0.f;
#pragma unroll
    for (int wv = 0; wv < 8; ++wv) sum += lds_part[wv * 256 + j * 32 + l];
    const int m = j + ((l >> 4) << 3);
    const int n = l & 15;
    lds_tile[m * 16 + n] = sum;
  }
  __syncthreads();

  // ---- stage 3: bias + log_softmax, one thread per batch row ----
  if (tid < 16) {
    const int b = row0 + tid;
    if (b < B) {
      float lg[10];
      float mx = -1e30f;
#pragma unroll
      for (int n = 0; n < 10; ++n) {
        lg[n] = lds_tile[tid * 16 + n] + fc_b[n];
        mx = fmaxf(mx, lg[n]);
      }
      float se = 0.f;
#pragma unroll
      for (int n = 0; n < 10; ++n) se += expf(lg[n] - mx);
      const float lse = mx + logf(se);
#pragma unroll
      for (int n = 0; n < 10; ++n) out[(size_t)b * 10 + n] = lg[n] - lse;
    }
  }
}

extern "C" void kernel_launch(void* const* d_in, const int* in_sizes, int n_in,
                              void* d_out, int out_size, void* d_ws, size_t ws_size,
                              hipStream_t stream) {
  const float* x      = (const float*)d_in[0];
  const float* conv_w = (const float*)d_in[1];
  const float* conv_b = (const float*)d_in[2];
  const float* angles = (const float*)d_in[3];
  const float* fc_w   = (const float*)d_in[4];
  const float* fc_b   = (const float*)d_in[5];
  float* out          = (float*)d_out;

  const int B = in_sizes[0] / 784;          // x is [B,1,28,28]
  const int grid = (B + 15) / 16;           // 16 batch rows per workgroup
  quanv_fused_kernel<<<grid, 256, 0, stream>>>(x, conv_w, conv_b, angles,
                                               fc_w, fc_b, out, B);
}